// DirVGAEModel_75814762709155
// MI455X (gfx1250) — compile-verified
//
#include <hip/hip_runtime.h>
#include <math.h>

// Problem constants (match the reference)
#define NN   12288
#define EE   393216
#define CIN  128
#define CHID 128
#define COUT 64
#define NEG_SLOPE 0.2f

typedef float v2f __attribute__((ext_vector_type(2)));
typedef float v8f __attribute__((ext_vector_type(8)));

// ---------- helpers ----------
__device__ __forceinline__ unsigned encf(float f) {
    unsigned u = __float_as_uint(f);
    return (u & 0x80000000u) ? ~u : (u | 0x80000000u);   // monotonic float->uint
}
__device__ __forceinline__ float decf(unsigned u) {
    unsigned v = (u & 0x80000000u) ? (u & 0x7FFFFFFFu) : ~u;
    return __uint_as_float(v);
}
__device__ __forceinline__ float fatomic_add(float* p, float v) {
    return __hip_atomic_fetch_add(p, v, __ATOMIC_RELAXED, __HIP_MEMORY_SCOPE_AGENT);
}
__device__ __forceinline__ unsigned long long splitmix(unsigned long long x) {
    x += 0x9E3779B97F4A7C15ull;
    x = (x ^ (x >> 30)) * 0xBF58476D1CE4E5B9ull;
    x = (x ^ (x >> 27)) * 0x94D049BB133111EBull;
    return x ^ (x >> 31);
}
__device__ __forceinline__ float u01(unsigned long long h) {
    return ((float)((h >> 40) & 0xFFFFFFull) + 1.0f) * (1.0f / 16777218.0f); // (0,1)
}

// ---------- WMMA fp32 GEMM: Y[N x C] = X[N x K] @ W[C x K]^T + bias ----------
// 8 waves/block; each wave computes one 16x16 tile with v_wmma_f32_16x16x4_f32.
__global__ __launch_bounds__(256)
void k_linear_wmma(const float* __restrict__ X, const float* __restrict__ W,
                   const float* __restrict__ bias, float* __restrict__ Y,
                   int K, int C) {
    const int wave = threadIdx.x >> 5;
    const int lane = threadIdx.x & 31;
    const int row0 = (blockIdx.x * 8 + wave) * 16;
    const int col0 = blockIdx.y * 16;
    const int half = lane >> 4, lid = lane & 15;

    const float* __restrict__ xr = X + (size_t)(row0 + lid) * K;  // A row (M = lid)
    const float* __restrict__ wr = W + (size_t)(col0 + lid) * K;  // B col (N = lid)
    v8f acc = {};
    for (int k = 0; k < K; k += 4) {
        v2f a, b;
        a.x = xr[k + 2 * half];     a.y = xr[k + 2 * half + 1];   // A: K = 2*half, 2*half+1
        b.x = wr[k + 2 * half];     b.y = wr[k + 2 * half + 1];   // B^T read = W row
        acc = __builtin_amdgcn_wmma_f32_16x16x4_f32(false, a, false, b,
                                                    (short)0, acc, false, false);
    }
    const float bb = bias[col0 + lid];
#pragma unroll
    for (int v = 0; v < 8; ++v)
        Y[(size_t)(row0 + half * 8 + v) * C + col0 + lid] = acc[v] + bb;
}

// ---------- init kernels ----------
__global__ void k_init_seg(unsigned* __restrict__ menc, float* __restrict__ ssum, int n) {
    int i = blockIdx.x * blockDim.x + threadIdx.x;
    if (i < n) { menc[i] = 0u; ssum[i] = 0.0f; }
}
__global__ void k_init_bias(float* __restrict__ out, const float* __restrict__ bias,
                            int n, int C) {
    int i = blockIdx.x * blockDim.x + threadIdx.x;
    if (i < n) out[i] = bias[i % C];
}

// ---------- edge logits + segment max (fused), wave per edge ----------
template <int C>
__global__ __launch_bounds__(256)
void k_edge_logits(const float* __restrict__ xl, const float* __restrict__ xr,
                   const float* __restrict__ att, const int* __restrict__ src,
                   const int* __restrict__ dst, float* __restrict__ elog,
                   unsigned* __restrict__ menc) {
    int gid = blockIdx.x * blockDim.x + threadIdx.x;
    int e = gid >> 5, lane = gid & 31;
    if (e >= EE) return;
    const int s = src[e], d = dst[e];
    float acc = 0.0f;
    if constexpr (C == 128) {
        const float4 a = *reinterpret_cast<const float4*>(xl + (size_t)s * C + lane * 4);
        const float4 b = *reinterpret_cast<const float4*>(xr + (size_t)d * C + lane * 4);
        const float4 t = *reinterpret_cast<const float4*>(att + lane * 4);
        float h;
        h = a.x + b.x; h = h > 0.f ? h : NEG_SLOPE * h; acc += h * t.x;
        h = a.y + b.y; h = h > 0.f ? h : NEG_SLOPE * h; acc += h * t.y;
        h = a.z + b.z; h = h > 0.f ? h : NEG_SLOPE * h; acc += h * t.z;
        h = a.w + b.w; h = h > 0.f ? h : NEG_SLOPE * h; acc += h * t.w;
    } else {
        const float2 a = *reinterpret_cast<const float2*>(xl + (size_t)s * C + lane * 2);
        const float2 b = *reinterpret_cast<const float2*>(xr + (size_t)d * C + lane * 2);
        const float2 t = *reinterpret_cast<const float2*>(att + lane * 2);
        float h;
        h = a.x + b.x; h = h > 0.f ? h : NEG_SLOPE * h; acc += h * t.x;
        h = a.y + b.y; h = h > 0.f ? h : NEG_SLOPE * h; acc += h * t.y;
    }
#pragma unroll
    for (int off = 16; off; off >>= 1) acc += __shfl_down(acc, off, 32);
    if (lane == 0) { elog[e] = acc; atomicMax(menc + d, encf(acc)); }
}

// ---------- exp(e - m[dst]) + segment sum ----------
__global__ __launch_bounds__(256)
void k_edge_expsum(const float* __restrict__ elog, const int* __restrict__ dst,
                   const unsigned* __restrict__ menc, float* __restrict__ aexp,
                   float* __restrict__ ssum) {
    int e = blockIdx.x * blockDim.x + threadIdx.x;
    if (e >= EE) return;
    const int d = dst[e];
    const float a = expf(elog[e] - decf(menc[d]));
    aexp[e] = a;
    fatomic_add(ssum + d, a);
}

// ---------- out[dst] += alpha * xl[src], wave per edge ----------
template <int C>
__global__ __launch_bounds__(256)
void k_edge_scatter(const float* __restrict__ xl, const int* __restrict__ src,
                    const int* __restrict__ dst, const float* __restrict__ aexp,
                    const float* __restrict__ ssum, float* __restrict__ out) {
    int gid = blockIdx.x * blockDim.x + threadIdx.x;
    int e = gid >> 5, lane = gid & 31;
    if (e >= EE) return;
    const int s = src[e], d = dst[e];
    const float w = aexp[e] / (ssum[d] + 1e-16f);
    if constexpr (C == 128) {
        const float4 a = *reinterpret_cast<const float4*>(xl + (size_t)s * C + lane * 4);
        float* o = out + (size_t)d * C + lane * 4;
        fatomic_add(o + 0, w * a.x); fatomic_add(o + 1, w * a.y);
        fatomic_add(o + 2, w * a.z); fatomic_add(o + 3, w * a.w);
    } else {
        const float2 a = *reinterpret_cast<const float2*>(xl + (size_t)s * C + lane * 2);
        float* o = out + (size_t)d * C + lane * 2;
        fatomic_add(o + 0, w * a.x); fatomic_add(o + 1, w * a.y);
    }
}

// ---------- activations ----------
__global__ void k_relu(float* __restrict__ x, int n) {
    int i = blockIdx.x * blockDim.x + threadIdx.x;
    if (i < n) x[i] = fmaxf(x[i], 0.0f);
}
__global__ void k_softplus_eps(float* __restrict__ x, int n) {
    int i = blockIdx.x * blockDim.x + threadIdx.x;
    if (i < n) {
        float v = x[i];
        v = (v > 20.0f) ? v : log1pf(expf(v));
        x[i] = v + 1e-6f;
    }
}

// ---------- Gamma(alpha,1) sampler: Marsaglia-Tsang with stateless hash RNG ----------
__global__ __launch_bounds__(256)
void k_gamma(const float* __restrict__ alpha, float* __restrict__ g, int n) {
    int i = blockIdx.x * blockDim.x + threadIdx.x;
    if (i >= n) return;
    const float al = alpha[i];
    const bool boost = (al < 1.0f);
    const float ap = boost ? al + 1.0f : al;
    const float d  = ap - (1.0f / 3.0f);
    const float cc = rsqrtf(9.0f * d);
    unsigned long long ctr = ((unsigned long long)i << 8) ^ 0x2A2A2A2A42ull;
    float gres = d;  // fallback (mean-ish) if all rejections fail
    for (int it = 0; it < 32; ++it) {
        const float u1 = u01(splitmix(ctr + 3ull * it + 0));
        const float u2 = u01(splitmix(ctr + 3ull * it + 1));
        const float r  = sqrtf(-2.0f * logf(u1));
        const float xN = r * cosf(6.2831853f * u2);           // Box-Muller normal
        float v = 1.0f + cc * xN;
        v = v * v * v;
        if (v <= 0.0f) continue;
        const float u = u01(splitmix(ctr + 3ull * it + 2));
        if (logf(u) < 0.5f * xN * xN + d - d * v + d * logf(v)) { gres = d * v; break; }
    }
    if (boost) gres *= powf(u01(splitmix(ctr + 777ull)), 1.0f / al);
    g[i] = fmaxf(gres, 1e-20f);
}

// ---------- z = g / rowsum(g), wave per node ----------
__global__ __launch_bounds__(256)
void k_znorm(const float* __restrict__ g, float* __restrict__ z) {
    int gid = blockIdx.x * blockDim.x + threadIdx.x;
    int node = gid >> 5, lane = gid & 31;
    if (node >= NN) return;
    const float a0 = g[(size_t)node * COUT + lane];
    const float a1 = g[(size_t)node * COUT + 32 + lane];
    float s = a0 + a1;
#pragma unroll
    for (int off = 16; off; off >>= 1) s += __shfl_xor(s, off, 32);
    const float inv = 1.0f / s;
    z[(size_t)node * COUT + lane]      = a0 * inv;
    z[(size_t)node * COUT + 32 + lane] = a1 * inv;
}

// ---------- adj = sigmoid(z @ z^T), WMMA fp32, nontemporal 604MB store ----------
__global__ __launch_bounds__(256)
void k_adj_wmma(const float* __restrict__ z, float* __restrict__ adj) {
    const int wave = threadIdx.x >> 5;
    const int lane = threadIdx.x & 31;
    const int row0 = (blockIdx.x * 8 + wave) * 16;
    const int col0 = blockIdx.y * 16;
    const int half = lane >> 4, lid = lane & 15;

    const float* __restrict__ ar = z + (size_t)(row0 + lid) * COUT;
    const float* __restrict__ br = z + (size_t)(col0 + lid) * COUT;
    v8f acc = {};
#pragma unroll
    for (int k = 0; k < COUT; k += 4) {
        v2f a, b;
        a.x = ar[k + 2 * half]; a.y = ar[k + 2 * half + 1];
        b.x = br[k + 2 * half]; b.y = br[k + 2 * half + 1];
        acc = __builtin_amdgcn_wmma_f32_16x16x4_f32(false, a, false, b,
                                                    (short)0, acc, false, false);
    }
#pragma unroll
    for (int v = 0; v < 8; ++v) {
        const float sg = 1.0f / (1.0f + expf(-acc[v]));
        // streaming store: adjacency is 604MB, keep it out of L2
        __builtin_nontemporal_store(sg, adj + (size_t)(row0 + half * 8 + v) * NN + col0 + lid);
    }
}

// ---------- host side ----------
extern "C" void kernel_launch(void* const* d_in, const int* in_sizes, int n_in,
                              void* d_out, int out_size, void* d_ws, size_t ws_size,
                              hipStream_t stream) {
    const float* x    = (const float*)d_in[0];
    const int*   edge = (const int*)d_in[1];   // [2, E] (JAX default int32)
    const int*   src  = edge;
    const int*   dst  = edge + EE;
    const float* Wl1 = (const float*)d_in[2];  const float* bl1 = (const float*)d_in[3];
    const float* Wr1 = (const float*)d_in[4];  const float* br1 = (const float*)d_in[5];
    const float* att1= (const float*)d_in[6];  const float* b1  = (const float*)d_in[7];
    const float* Wl2 = (const float*)d_in[8];  const float* bl2 = (const float*)d_in[9];
    const float* Wr2 = (const float*)d_in[10]; const float* br2 = (const float*)d_in[11];
    const float* att2= (const float*)d_in[12]; const float* b2  = (const float*)d_in[13];

    // output layout: adj [N*N], alpha [N*64], z [N*64]
    float* adj       = (float*)d_out;
    float* alpha_out = adj + (size_t)NN * NN;
    float* z_out     = alpha_out + (size_t)NN * COUT;

    // workspace carve-up (~31.6 MB)
    float*    xl1  = (float*)d_ws;            // N*128
    float*    xr1  = xl1 + (size_t)NN * CHID; // N*128
    float*    h    = xr1 + (size_t)NN * CHID; // N*128
    float*    e1   = h   + (size_t)NN * CHID; // E
    float*    aexp = e1  + EE;                // E
    unsigned* menc = (unsigned*)(aexp + EE);  // N
    float*    ssum = (float*)(menc + NN);     // N
    float*    xl2  = ssum + NN;               // N*64
    float*    xr2  = xl2 + (size_t)NN * COUT; // N*64
    float*    g    = xr2 + (size_t)NN * COUT; // N*64

    const int TPB = 256;
    const int edgeWaveBlocks = (EE * 32) / TPB;  // wave-per-edge kernels
    const int edgeBlocks     = EE / TPB;

    // ---- conv1 (IN=128 -> HID=128) ----
    k_linear_wmma<<<dim3(NN / 128, CHID / 16), TPB, 0, stream>>>(x, Wl1, bl1, xl1, CIN, CHID);
    k_linear_wmma<<<dim3(NN / 128, CHID / 16), TPB, 0, stream>>>(x, Wr1, br1, xr1, CIN, CHID);
    k_init_seg<<<(NN + TPB - 1) / TPB, TPB, 0, stream>>>(menc, ssum, NN);
    k_init_bias<<<(NN * CHID) / TPB, TPB, 0, stream>>>(h, b1, NN * CHID, CHID);
    k_edge_logits<CHID><<<edgeWaveBlocks, TPB, 0, stream>>>(xl1, xr1, att1, src, dst, e1, menc);
    k_edge_expsum<<<edgeBlocks, TPB, 0, stream>>>(e1, dst, menc, aexp, ssum);
    k_edge_scatter<CHID><<<edgeWaveBlocks, TPB, 0, stream>>>(xl1, src, dst, aexp, ssum, h);
    k_relu<<<(NN * CHID) / TPB, TPB, 0, stream>>>(h, NN * CHID);

    // ---- conv2 (HID=128 -> OUT=64), accumulate straight into alpha output ----
    k_linear_wmma<<<dim3(NN / 128, COUT / 16), TPB, 0, stream>>>(h, Wl2, bl2, xl2, CHID, COUT);
    k_linear_wmma<<<dim3(NN / 128, COUT / 16), TPB, 0, stream>>>(h, Wr2, br2, xr2, CHID, COUT);
    k_init_seg<<<(NN + TPB - 1) / TPB, TPB, 0, stream>>>(menc, ssum, NN);
    k_init_bias<<<(NN * COUT) / TPB, TPB, 0, stream>>>(alpha_out, b2, NN * COUT, COUT);
    k_edge_logits<COUT><<<edgeWaveBlocks, TPB, 0, stream>>>(xl2, xr2, att2, src, dst, e1, menc);
    k_edge_expsum<<<edgeBlocks, TPB, 0, stream>>>(e1, dst, menc, aexp, ssum);
    k_edge_scatter<COUT><<<edgeWaveBlocks, TPB, 0, stream>>>(xl2, src, dst, aexp, ssum, alpha_out);
    k_softplus_eps<<<(NN * COUT) / TPB, TPB, 0, stream>>>(alpha_out, NN * COUT);

    // ---- Dirichlet-like latent ----
    k_gamma<<<(NN * COUT) / TPB, TPB, 0, stream>>>(alpha_out, g, NN * COUT);
    k_znorm<<<(NN * 32) / TPB, TPB, 0, stream>>>(g, z_out);

    // ---- adjacency: sigmoid(z z^T), 768x768 tiles of 16x16 ----
    k_adj_wmma<<<dim3(NN / 128, NN / 16), TPB, 0, stream>>>(z_out, adj);
}